// ALIGNNlayer_62311385530743
// MI455X (gfx1250) — compile-verified
//
#include <hip/hip_runtime.h>
#include <hip/hip_bf16.h>

typedef float v2f __attribute__((ext_vector_type(2)));
typedef float v8f __attribute__((ext_vector_type(8)));

#define NND 100000
#define NED 600000
#define NAN_ 600000
#define DD  128

// ---------------------------------------------------------------------------
// Tiny one-shot transpose: Wt[m][n][k] = W[m][k][n]  (nmat 128x128 matrices)
// Makes B fragments contiguous (fixed N, consecutive K -> one b64 load).
// ---------------------------------------------------------------------------
__global__ __launch_bounds__(256) void transpose_w(const float* __restrict__ W,
                                                   float* __restrict__ Wt, int nmat) {
  const long gid = (long)blockIdx.x * 256 + threadIdx.x;
  if (gid >= (long)nmat * DD * DD) return;
  const long m = gid >> 14;
  const long n = (gid >> 7) & 127;
  const long k = gid & 127;
  Wt[gid] = W[m * DD * DD + k * DD + n];
}

// ---------------------------------------------------------------------------
// GEMM: Y[rows,128] = X[rows,128] @ W[128,128] + bias[128]   (Wt = W^T)
// Block = 128 threads = 4 waves. Block covers 64 rows x 128 cols.
// Each wave: 4 row-tiles x 2 col-tiles = 8 accumulators (64 VGPRs).
// Per k-step: 4 b64 A-loads + 2 b64 B-loads feed 8 v_wmma_f32_16x16x4_f32.
// Epilogue: uniform full-block test -> unguarded clause-grouped stores on the
// fast path; per-row guards only in the single tail block.
// ---------------------------------------------------------------------------
__global__ __launch_bounds__(128) void gemm128_wmma_f32(
    const float* __restrict__ X, const float* __restrict__ Wt,
    const float* __restrict__ bias, float* __restrict__ Y, int rows) {
  const int wave = threadIdx.x >> 5;   // 0..3
  const int lane = threadIdx.x & 31;
  const int half = lane >> 4;          // selects K pair within the 4-wide step
  const int l15  = lane & 15;
  const long rb  = (long)blockIdx.x * 64;
  const int  n0  = wave * 32 + l15;
  const int  n1  = n0 + 16;

  const float* xr[4];
#pragma unroll
  for (int r = 0; r < 4; ++r) {
    long row = rb + r * 16 + l15;
    if (row >= rows) row = rows - 1;   // clamp tail loads (stores are guarded)
    xr[r] = X + row * (long)DD;
  }
  const float* wt0 = Wt + (long)n0 * DD;
  const float* wt1 = Wt + (long)n1 * DD;

  v8f c0[4], c1[4];
#pragma unroll
  for (int r = 0; r < 4; ++r) { c0[r] = {}; c1[r] = {}; }

#pragma unroll 2
  for (int k0 = 0; k0 < DD; k0 += 4) {
    const int kk = k0 + half * 2;                 // even -> 8B aligned
    const v2f b0 = *(const v2f*)(wt0 + kk);       // B: N=n0, K=kk,kk+1
    const v2f b1 = *(const v2f*)(wt1 + kk);
#pragma unroll
    for (int r = 0; r < 4; ++r) {
      const v2f a = *(const v2f*)(xr[r] + kk);    // A: M=l15, K=kk,kk+1
      c0[r] = __builtin_amdgcn_wmma_f32_16x16x4_f32(false, a, false, b0,
                                                    (short)0, c0[r], false, false);
      c1[r] = __builtin_amdgcn_wmma_f32_16x16x4_f32(false, a, false, b1,
                                                    (short)0, c1[r], false, false);
    }
  }
  const float bb0 = bias[n0];
  const float bb1 = bias[n1];

  if (rb + 64 <= rows) {
    // Fast path: full 64-row block, no guards -> stores can clause.
    float* yb = Y + (rb + half * 8) * (long)DD;
#pragma unroll
    for (int r = 0; r < 4; ++r) {
      float* yr = yb + (long)r * 16 * DD;
#pragma unroll
      for (int i = 0; i < 8; ++i) {
        yr[(long)i * DD + n0] = c0[r][i] + bb0;
        yr[(long)i * DD + n1] = c1[r][i] + bb1;
      }
    }
  } else {
    // Tail block (only the last node-projection block takes this path).
#pragma unroll
    for (int r = 0; r < 4; ++r) {
      const long rowb = rb + r * 16 + half * 8;   // C/D layout: VGPR i -> M=i+8*half
#pragma unroll
      for (int i = 0; i < 8; ++i) {
        const long row = rowb + i;
        if (row < rows) {
          Y[row * DD + n0] = c0[r][i] + bb0;
          Y[row * DD + n1] = c1[r][i] + bb1;
        }
      }
    }
  }
}

// ---------------------------------------------------------------------------
// Edge/gate kernel: e_new = Ce + Dh[src] + Eh[dst]; sigma = sigmoid(e_new);
// atomic: num[dst] += sigma*Bh[src]; den[dst] += sigma; e_io <- relu(e_new).
// 32 threads per edge, float4 per thread.
// ---------------------------------------------------------------------------
__global__ __launch_bounds__(256) void gated_edge_kernel(
    float* __restrict__ e_io, const float* __restrict__ Dh,
    const float* __restrict__ Eh, const float* __restrict__ Bh,
    const int* __restrict__ src, const int* __restrict__ dst,
    float* __restrict__ num, float* __restrict__ den, int nE) {
  const long gid = (long)blockIdx.x * 256 + threadIdx.x;
  const long e   = gid >> 5;
  if (e >= nE) return;
  const int  c = (int)(gid & 31) * 4;
  const long s = src[e];
  const long t = dst[e];

  float ce[4], dh[4], eh[4], bh[4], eo[4];
  *(float4*)ce = *(const float4*)(e_io + e * DD + c);
  *(float4*)dh = *(const float4*)(Dh + s * DD + c);
  *(float4*)eh = *(const float4*)(Eh + t * DD + c);
  *(float4*)bh = *(const float4*)(Bh + s * DD + c);
#pragma unroll
  for (int j = 0; j < 4; ++j) {
    const float en  = ce[j] + dh[j] + eh[j];
    const float sig = 1.0f / (1.0f + __expf(-en));   // v_exp_f32 path
    atomicAdd(num + t * DD + c + j, sig * bh[j]);
    atomicAdd(den + t * DD + c + j, sig);
    eo[j] = fmaxf(en, 0.0f);
  }
  *(float4*)(e_io + e * DD + c) = *(const float4*)eo;
}

// ---------------------------------------------------------------------------
// Node finalize: h_io <- relu(Ah + num/(den+eps)), float4 per thread.
// ---------------------------------------------------------------------------
__global__ __launch_bounds__(256) void gated_node_kernel(
    float* __restrict__ h_io, const float* __restrict__ num,
    const float* __restrict__ den, long total) {
  const long idx = ((long)blockIdx.x * 256 + threadIdx.x) * 4;
  if (idx >= total) return;
  float a[4], n[4], d[4], o[4];
  *(float4*)a = *(const float4*)(h_io + idx);
  *(float4*)n = *(const float4*)(num + idx);
  *(float4*)d = *(const float4*)(den + idx);
#pragma unroll
  for (int j = 0; j < 4; ++j)
    o[j] = fmaxf(a[j] + n[j] / (d[j] + 1e-6f), 0.0f);
  *(float4*)(h_io + idx) = *(const float4*)o;
}

extern "C" void kernel_launch(void* const* d_in, const int* in_sizes, int n_in,
                              void* d_out, int out_size, void* d_ws, size_t ws_size,
                              hipStream_t stream) {
  const float* node  = (const float*)d_in[0];
  const float* edge  = (const float*)d_in[1];
  const float* angle = (const float*)d_in[2];
  const int*   gei   = (const int*)d_in[3];  // [2, E]: row0 src, row1 dst
  const int*   lei   = (const int*)d_in[4];  // [2, A]
  const float* w1    = (const float*)d_in[5];
  const float* b1    = (const float*)d_in[6];
  const float* w2    = (const float*)d_in[7];
  const float* b2    = (const float*)d_in[8];

  const long N = NND, E = NED, A = NAN_;
  float* h_out = (float*)d_out;                  // [N,128]
  float* e_out = h_out + N * DD;                 // [E,128]
  float* a_out = e_out + (long)E * DD;           // [A,128]

  const long eb = (long)E * DD;                  // floats per edge-sized buffer
  float* buf0 = (float*)d_ws;                    // Ce / relu(e_new1) (pass-2 input)
  float* buf1 = buf0 + eb;                       // Bh  / Bh2
  float* buf2 = buf1 + eb;                       // Dh  / Dh2
  float* buf3 = buf2 + eb;                       // Eh  / Eh2
  float* buf4 = buf3 + eb;                       // num / num2
  float* buf5 = buf4 + eb;                       // den / den2
  float* wt1  = buf5 + eb;                       // W1^T [5,128,128]
  float* wt2  = wt1 + 5 * DD * DD;               // W2^T [5,128,128]

  const int WMAT = DD * DD;  // 16384 floats per projection matrix

  // One-shot weight transposes (1.3 MB total, negligible).
  transpose_w<<<dim3((5 * WMAT + 255) / 256), dim3(256), 0, stream>>>(w1, wt1, 5);
  transpose_w<<<dim3((5 * WMAT + 255) / 256), dim3(256), 0, stream>>>(w2, wt2, 5);

  const unsigned gN = (unsigned)((N + 63) / 64);
  const unsigned gE = (unsigned)((E + 63) / 64);
  const unsigned gA = (unsigned)((A + 63) / 64);

  // ---------------- Pass 1: atom graph ----------------
  gemm128_wmma_f32<<<dim3(gN), dim3(128), 0, stream>>>(node, wt1 + 0 * WMAT, b1 + 0 * DD, h_out, (int)N); // Ah
  gemm128_wmma_f32<<<dim3(gN), dim3(128), 0, stream>>>(node, wt1 + 1 * WMAT, b1 + 1 * DD, buf1,  (int)N); // Bh
  gemm128_wmma_f32<<<dim3(gN), dim3(128), 0, stream>>>(node, wt1 + 3 * WMAT, b1 + 3 * DD, buf2,  (int)N); // Dh
  gemm128_wmma_f32<<<dim3(gN), dim3(128), 0, stream>>>(node, wt1 + 4 * WMAT, b1 + 4 * DD, buf3,  (int)N); // Eh
  gemm128_wmma_f32<<<dim3(gE), dim3(128), 0, stream>>>(edge, wt1 + 2 * WMAT, b1 + 2 * DD, buf0,  (int)E); // Ce

  hipMemsetAsync(buf4, 0, (size_t)N * DD * sizeof(float), stream);
  hipMemsetAsync(buf5, 0, (size_t)N * DD * sizeof(float), stream);

  gated_edge_kernel<<<dim3((unsigned)((E * 32 + 255) / 256)), dim3(256), 0, stream>>>(
      buf0, buf2, buf3, buf1, gei, gei + E, buf4, buf5, (int)E);
  gated_node_kernel<<<dim3((unsigned)((N * DD / 4 + 255) / 256)), dim3(256), 0, stream>>>(
      h_out, buf4, buf5, N * (long)DD);

  // ---------------- Pass 2: line graph (nodes = edges, edges = angles) -----
  gemm128_wmma_f32<<<dim3(gE), dim3(128), 0, stream>>>(buf0,  wt2 + 0 * WMAT, b2 + 0 * DD, e_out, (int)E); // Ah2
  gemm128_wmma_f32<<<dim3(gE), dim3(128), 0, stream>>>(buf0,  wt2 + 1 * WMAT, b2 + 1 * DD, buf1,  (int)E); // Bh2
  gemm128_wmma_f32<<<dim3(gE), dim3(128), 0, stream>>>(buf0,  wt2 + 3 * WMAT, b2 + 3 * DD, buf2,  (int)E); // Dh2
  gemm128_wmma_f32<<<dim3(gE), dim3(128), 0, stream>>>(buf0,  wt2 + 4 * WMAT, b2 + 4 * DD, buf3,  (int)E); // Eh2
  gemm128_wmma_f32<<<dim3(gA), dim3(128), 0, stream>>>(angle, wt2 + 2 * WMAT, b2 + 2 * DD, a_out, (int)A); // Ca

  hipMemsetAsync(buf4, 0, (size_t)E * DD * sizeof(float), stream);
  hipMemsetAsync(buf5, 0, (size_t)E * DD * sizeof(float), stream);

  gated_edge_kernel<<<dim3((unsigned)((A * 32 + 255) / 256)), dim3(256), 0, stream>>>(
      a_out, buf2, buf3, buf1, lei, lei + A, buf4, buf5, (int)A);
  gated_node_kernel<<<dim3((unsigned)((E * DD / 4 + 255) / 256)), dim3(256), 0, stream>>>(
      e_out, buf4, buf5, E * (long)DD);
}